// SR_SIMLoss_39754217292402
// MI455X (gfx1250) — compile-verified
//
#include <hip/hip_runtime.h>

typedef __attribute__((ext_vector_type(2))) float v2f;
typedef __attribute__((ext_vector_type(8))) float v8f;

#define PI_F 3.14159265358979323846f

static __device__ __forceinline__ int reflect2048(int i) {
  i = (i < 0) ? -i : i;
  return (i > 2047) ? 4094 - i : i;
}
static __device__ __forceinline__ int reflect64(int i) {
  i = (i < 0) ? -i : i;
  return (i > 63) ? 126 - i : i;
}
static __device__ __forceinline__ int clamp64(int i) {
  return i < 0 ? 0 : (i > 63 ? 63 : i);
}

// ---------------------------------------------------------------------------
// Stage 1: fused reflect-pad 8x8 average downsample + RGB->YIQ for both images
// Reads each input element exactly once (bandwidth-bound, ~100MB total).
// grid = 256 blocks (row), block = 256 threads (col)
// ---------------------------------------------------------------------------
__global__ __launch_bounds__(256) void k_down_yiq(
    const float* __restrict__ imx, const float* __restrict__ imy,
    float* __restrict__ Y1, float* __restrict__ I1, float* __restrict__ Q1,
    float* __restrict__ Y2, float* __restrict__ I2, float* __restrict__ Q2)
{
  const int j = threadIdx.x;
  const int i = blockIdx.x;
  const int o = i * 256 + j;
  #pragma unroll
  for (int im = 0; im < 2; ++im) {
    const float* __restrict__ p = im ? imy : imx;
    float r = 0.f, g = 0.f, b = 0.f;
    #pragma unroll
    for (int u = 0; u < 8; ++u) {
      const int ri = reflect2048(i * 8 + u - 3);
      const float* __restrict__ row = p + (size_t)ri * 6144;  // 2048*3
      #pragma unroll
      for (int v = 0; v < 8; ++v) {
        const int ci = reflect2048(j * 8 + v - 3);
        const float* __restrict__ px = row + ci * 3;
        r += px[0]; g += px[1]; b += px[2];
      }
    }
    r *= 0.015625f; g *= 0.015625f; b *= 0.015625f;
    const float Yv = 0.299f * r + 0.587f * g + 0.114f * b;
    const float Iv = 0.5959f * r - 0.2746f * g - 0.3213f * b;
    const float Qv = 0.2115f * r - 0.5227f * g + 0.3112f * b;
    if (im == 0) { Y1[o] = Yv; I1[o] = Iv; Q1[o] = Qv; }
    else         { Y2[o] = Yv; I2[o] = Iv; Q2[o] = Qv; }
  }
}

// ---------------------------------------------------------------------------
// WMMA f32 GEMM: C = alpha*A*(B or B^T) (+ C if ACC)
// Template params make the inner loop branch-free: clean load -> mul -> wmma.
// One wave (32 threads) per 16x16 output tile; K stepped by 4 through
// V_WMMA_F32_16X16X4_F32, unrolled x4. M,N multiples of 16; K multiple of 16.
// A frag (16x4): lanes 0-15 -> rows, VGPR0/1 = K{0,1}; lanes 16-31 = K{2,3}.
// B frag (4x16): lanes index N, half selects K pair. C/D per ISA layout.
// ---------------------------------------------------------------------------
template <int TRANSB, int ACC>
__global__ __launch_bounds__(32) void k_wmma_gemm(
    const float* __restrict__ A, int lda,
    const float* __restrict__ B, int ldb,
    float* __restrict__ C, int ldc,
    int N, int K, float alpha)
{
  const int tilesN = N >> 4;
  const int tm = (blockIdx.x / tilesN) << 4;
  const int tn = (blockIdx.x % tilesN) << 4;
  const int lane = threadIdx.x;
  const int half = lane >> 4;    // 0 or 1
  const int l16  = lane & 15;

  const float* __restrict__ Arow = A + (size_t)(tm + l16) * lda + 2 * half;
  const float* __restrict__ Bp =
      TRANSB ? (B + (size_t)(tn + l16) * ldb + 2 * half)
             : (B + (size_t)(2 * half) * ldb + (tn + l16));

  v8f acc;
  if (ACC) {
    #pragma unroll
    for (int r = 0; r < 8; ++r)
      acc[r] = C[(size_t)(tm + r + 8 * half) * ldc + (tn + l16)];
  } else {
    #pragma unroll
    for (int r = 0; r < 8; ++r) acc[r] = 0.f;
  }

  #pragma unroll 4
  for (int k = 0; k < K; k += 4) {
    v2f a, b;
    a.x = alpha * Arow[k + 0];
    a.y = alpha * Arow[k + 1];
    if (TRANSB) {
      b.x = Bp[k + 0];
      b.y = Bp[k + 1];
    } else {
      b.x = Bp[(size_t)(k + 0) * ldb];
      b.y = Bp[(size_t)(k + 1) * ldb];
    }
    acc = __builtin_amdgcn_wmma_f32_16x16x4_f32(
        false, a, false, b, (short)0, acc, false, false);
  }

  #pragma unroll
  for (int r = 0; r < 8; ++r)
    C[(size_t)(tm + r + 8 * half) * ldc + (tn + l16)] = acc[r];
}

// ---------------------------------------------------------------------------
// Constant-operator fill kernels
// ---------------------------------------------------------------------------
// DFT matrix F[j][k] = exp(-2*pi*i*j*k/64). grid 16 x 256.
__global__ __launch_bounds__(256) void k_fill_dft(float* __restrict__ Fr,
                                                  float* __restrict__ Fi)
{
  const int idx = blockIdx.x * 256 + threadIdx.x;   // 0..4095
  const int j = idx >> 6, k = idx & 63;
  const int m = (j * k) & 63;
  const float t = -(PI_F / 32.0f) * (float)m;
  Fr[idx] = cosf(t);
  Fi[idx] = sinf(t);
}

// Keys cubic (a=-0.5) antialiased 256->64 resize weights, row-normalized
// like jax.image.resize. grid 64 blocks (output row), block 256 (input col).
__global__ __launch_bounds__(256) void k_fill_cubic(float* __restrict__ A)
{
  __shared__ float sm[256];
  const int i = blockIdx.x, j = threadIdx.x;
  const float center = (i + 0.5f) * 4.0f - 0.5f;
  const float x = fabsf(((float)j - center) * 0.25f);
  float w = 0.f;
  if (x < 1.0f)      w = (1.5f * x - 2.5f) * x * x + 1.0f;
  else if (x < 2.0f) w = -0.5f * (((x - 5.0f) * x + 8.0f) * x - 4.0f);
  sm[j] = w;
  __syncthreads();
  for (int s = 128; s > 0; s >>= 1) {
    if (j < s) sm[j] += sm[j + s];
    __syncthreads();
  }
  A[i * 256 + j] = w / sm[0];
}

// Linear 64->256 upsample weights, row-normalized. grid 256, block 64.
__global__ __launch_bounds__(64) void k_fill_linear(float* __restrict__ L)
{
  __shared__ float sm[64];
  const int i = blockIdx.x, j = threadIdx.x;
  const float src = (i + 0.5f) * 0.25f - 0.5f;
  float w = fmaxf(0.f, 1.f - fabsf((float)j - src));
  sm[j] = w;
  __syncthreads();
  for (int s = 32; s > 0; s >>= 1) {
    if (j < s) sm[j] += sm[j + s];
    __syncthreads();
  }
  L[i * 64 + j] = w / sm[0];
}

// Normalized 9-tap Gaussian, sigma = 3.8. <<<1,1>>>
__global__ void k_fill_gauss9(float* __restrict__ g)
{
  float tmp[9], s = 0.f;
  for (int t = 0; t < 9; ++t) {
    const float x = (float)(t - 4);
    tmp[t] = expf(-(x * x) / (2.0f * 3.8f * 3.8f));
    s += tmp[t];
  }
  for (int t = 0; t < 9; ++t) g[t] = tmp[t] / s;
}

// ---------------------------------------------------------------------------
// Spectral-residual elementwise kernels (all on 64x64 = 4096 elems, <<<16,256>>>)
// ---------------------------------------------------------------------------
__global__ __launch_bounds__(256) void k_logamp_phase(
    const float* __restrict__ Xr, const float* __restrict__ Xi,
    float* __restrict__ la, float* __restrict__ ph)
{
  const int idx = blockIdx.x * 256 + threadIdx.x;
  const float xr = Xr[idx], xi = Xi[idx];
  la[idx] = 0.5f * logf(fmaxf(xr * xr + xi * xi, 1e-45f));
  ph[idx] = atan2f(xi, xr);
}

__global__ __launch_bounds__(256) void k_box3(const float* __restrict__ in,
                                              float* __restrict__ out)
{
  const int idx = blockIdx.x * 256 + threadIdx.x;
  const int i = idx >> 6, j = idx & 63;
  float s = 0.f;
  #pragma unroll
  for (int di = -1; di <= 1; ++di)
    #pragma unroll
    for (int dj = -1; dj <= 1; ++dj)
      s += in[clamp64(i + di) * 64 + clamp64(j + dj)];
  out[idx] = s * (1.0f / 9.0f);
}

__global__ __launch_bounds__(256) void k_spectral_exp(
    const float* __restrict__ la, const float* __restrict__ bx,
    const float* __restrict__ ph,
    float* __restrict__ Rr, float* __restrict__ Ri)
{
  const int idx = blockIdx.x * 256 + threadIdx.x;
  const float e = expf(la[idx] - bx[idx]);
  const float p = ph[idx];
  Rr[idx] = e * cosf(p);
  Ri[idx] = e * sinf(p);
}

// sal = |ifft|^2, with the 1/(64*64) ifft scale squared folded in.
__global__ __launch_bounds__(256) void k_mag2(const float* __restrict__ Ur,
                                              const float* __restrict__ Ui,
                                              float* __restrict__ sal)
{
  const int idx = blockIdx.x * 256 + threadIdx.x;
  const float ur = Ur[idx], ui = Ui[idx];
  sal[idx] = (ur * ur + ui * ui) * (1.0f / 16777216.0f);
}

__global__ __launch_bounds__(256) void k_gauss_blur(
    const float* __restrict__ in, const float* __restrict__ g9,
    float* __restrict__ out)
{
  const int idx = blockIdx.x * 256 + threadIdx.x;
  const int i = idx >> 6, j = idx & 63;
  float g[9];
  #pragma unroll
  for (int t = 0; t < 9; ++t) g[t] = g9[t];
  float s = 0.f;
  #pragma unroll
  for (int u = 0; u < 9; ++u) {
    const int ri = reflect64(i + u - 4);
    float rs = 0.f;
    #pragma unroll
    for (int v = 0; v < 9; ++v)
      rs += g[v] * in[ri * 64 + reflect64(j + v - 4)];
    s += g[u] * rs;
  }
  out[idx] = s;
}

// Per-row L2 normalization (axis=1). grid 64 (row), block 64 (col).
__global__ __launch_bounds__(64) void k_rownorm(const float* __restrict__ in,
                                                float* __restrict__ out)
{
  __shared__ float sm[64];
  const int i = blockIdx.x, j = threadIdx.x;
  const float v = in[i * 64 + j];
  sm[j] = v * v;
  __syncthreads();
  for (int s = 32; s > 0; s >>= 1) {
    if (j < s) sm[j] += sm[j + s];
    __syncthreads();
  }
  out[i * 64 + j] = v / fmaxf(sqrtf(sm[0]), 1e-12f);
}

// ---------------------------------------------------------------------------
// Scharr-style gradient magnitude, true convolution with zero pad (256x256).
// grid 256 (row), block 256 (col)
// ---------------------------------------------------------------------------
__global__ __launch_bounds__(256) void k_grad(const float* __restrict__ img,
                                              float* __restrict__ g)
{
  const int i = blockIdx.x, j = threadIdx.x;
  const float kx[3][3] = {{ 3.f/16.f, 0.f, -3.f/16.f},
                          {10.f/16.f, 0.f, -10.f/16.f},
                          { 3.f/16.f, 0.f, -3.f/16.f}};
  float cx = 0.f, cy = 0.f;
  #pragma unroll
  for (int a = 0; a < 3; ++a)
    #pragma unroll
    for (int b = 0; b < 3; ++b) {
      const int ii = i + 1 - a, jj = j + 1 - b;
      const float v = (ii >= 0 && ii < 256 && jj >= 0 && jj < 256)
                          ? img[ii * 256 + jj] : 0.f;
      cx += kx[a][b] * v;
      cy += kx[b][a] * v;   // dy = dx^T
    }
  g[i * 256 + j] = sqrtf(cx * cx + cy * cy);
}

// ---------------------------------------------------------------------------
// Final SR-SIM combine + deterministic two-stage reduction
// ---------------------------------------------------------------------------
__global__ __launch_bounds__(256) void k_combine(
    const float* __restrict__ S1, const float* __restrict__ S2,
    const float* __restrict__ G1, const float* __restrict__ G2,
    const float* __restrict__ I1, const float* __restrict__ I2,
    const float* __restrict__ Q1, const float* __restrict__ Q2,
    float* __restrict__ PN, float* __restrict__ PD)
{
  __shared__ float sn[256], sd[256];
  const int t = threadIdx.x;
  const int idx = blockIdx.x * 256 + t;
  const float s1 = S1[idx], s2 = S2[idx];
  const float g1 = G1[idx], g2 = G2[idx];
  const float i1 = I1[idx], i2 = I2[idx];
  const float q1 = Q1[idx], q2 = Q2[idx];

  const float sal_sim  = (2.f * s1 * s2 + 0.4f)  / (s1 * s1 + s2 * s2 + 0.4f);
  const float grad_sim = (2.f * g1 * g2 + 225.f) / (g1 * g1 + g2 * g2 + 225.f);
  const float Rm  = fmaxf(s1, s2);
  const float sim = sal_sim * sqrtf(grad_sim);
  const float i_sim = (2.f * i1 * i2 + 200.f) / (i1 * i1 + i2 * i2 + 200.f);
  const float q_sim = (2.f * q1 * q2 + 200.f) / (q1 * q1 + q2 * q2 + 200.f);
  const float prod = i_sim * q_sim;
  const float sgn = (prod > 0.f) ? 1.f : ((prod < 0.f) ? -1.f : 0.f);
  const float sim_c = sim * sgn * powf(fabsf(prod), 0.03f);

  sn[t] = sim_c * Rm;
  sd[t] = Rm;
  __syncthreads();
  for (int s = 128; s > 0; s >>= 1) {
    if (t < s) { sn[t] += sn[t + s]; sd[t] += sd[t + s]; }
    __syncthreads();
  }
  if (t == 0) { PN[blockIdx.x] = sn[0]; PD[blockIdx.x] = sd[0]; }
}

__global__ __launch_bounds__(256) void k_final(const float* __restrict__ PN,
                                               const float* __restrict__ PD,
                                               float* __restrict__ out)
{
  __shared__ float sn[256], sd[256];
  const int t = threadIdx.x;
  sn[t] = PN[t];
  sd[t] = PD[t];
  __syncthreads();
  for (int s = 128; s > 0; s >>= 1) {
    if (t < s) { sn[t] += sn[t + s]; sd[t] += sd[t + s]; }
    __syncthreads();
  }
  if (t == 0) out[0] = 1.0f - sn[0] / sd[0];
}

// ---------------------------------------------------------------------------
// Host-side orchestration
// ---------------------------------------------------------------------------
static inline void gemm(hipStream_t st, const float* A, int lda,
                        const float* B, int ldb, float* C, int ldc,
                        int M, int N, int K, float alpha, int transB, int acc)
{
  const dim3 g((M >> 4) * (N >> 4));
  if (transB) {
    if (acc) k_wmma_gemm<1, 1><<<g, 32, 0, st>>>(A, lda, B, ldb, C, ldc, N, K, alpha);
    else     k_wmma_gemm<1, 0><<<g, 32, 0, st>>>(A, lda, B, ldb, C, ldc, N, K, alpha);
  } else {
    if (acc) k_wmma_gemm<0, 1><<<g, 32, 0, st>>>(A, lda, B, ldb, C, ldc, N, K, alpha);
    else     k_wmma_gemm<0, 0><<<g, 32, 0, st>>>(A, lda, B, ldb, C, ldc, N, K, alpha);
  }
}

// Full saliency pipeline for one 256x256 Y map -> 256x256 saliency map.
static void run_saliency(hipStream_t st, const float* Yin, float* Sout,
                         const float* ACUB, const float* LLIN,
                         const float* FR, const float* FI, const float* G9,
                         float* scr)
{
  float* T0   = scr;            // 64x256
  float* IN64 = T0 + 16384;     // 64x64
  float* TA   = IN64 + 4096;
  float* TB   = TA + 4096;
  float* XR   = TB + 4096;
  float* XI   = XR + 4096;
  float* LA   = XI + 4096;
  float* BX   = LA + 4096;
  float* PH   = BX + 4096;
  float* RR   = PH + 4096;
  float* RI   = RR + 4096;
  float* PR   = RI + 4096;
  float* PI_  = PR + 4096;
  float* UR   = PI_ + 4096;
  float* UI   = UR + 4096;
  float* SAL  = UI + 4096;
  float* BLR  = SAL + 4096;
  float* NRM  = BLR + 4096;
  float* UPT  = NRM + 4096;     // 256x64

  // cubic antialiased resize 256->64 : IN64 = ACUB * Yin * ACUB^T
  gemm(st, ACUB, 256, Yin, 256, T0, 256, 64, 256, 256, 1.f, 0, 0);
  gemm(st, T0, 256, ACUB, 256, IN64, 64, 64, 64, 256, 1.f, 1, 0);

  // fft2 via DFT matmul (F symmetric): X = F * in * F
  gemm(st, FR, 64, IN64, 64, TA, 64, 64, 64, 64, 1.f, 0, 0);
  gemm(st, FI, 64, IN64, 64, TB, 64, 64, 64, 64, 1.f, 0, 0);
  gemm(st, TA, 64, FR, 64, XR, 64, 64, 64, 64, 1.f, 0, 0);
  gemm(st, TB, 64, FI, 64, XR, 64, 64, 64, 64, -1.f, 0, 1);
  gemm(st, TA, 64, FI, 64, XI, 64, 64, 64, 64, 1.f, 0, 0);
  gemm(st, TB, 64, FR, 64, XI, 64, 64, 64, 64, 1.f, 0, 1);

  // spectral residual
  k_logamp_phase<<<16, 256, 0, st>>>(XR, XI, LA, PH);
  k_box3<<<16, 256, 0, st>>>(LA, BX);
  k_spectral_exp<<<16, 256, 0, st>>>(LA, BX, PH, RR, RI);

  // ifft2 via conj(F) matmuls (scale folded into k_mag2)
  gemm(st, FR, 64, RR, 64, PR, 64, 64, 64, 64, 1.f, 0, 0);
  gemm(st, FI, 64, RI, 64, PR, 64, 64, 64, 64, 1.f, 0, 1);
  gemm(st, FR, 64, RI, 64, PI_, 64, 64, 64, 64, 1.f, 0, 0);
  gemm(st, FI, 64, RR, 64, PI_, 64, 64, 64, 64, -1.f, 0, 1);
  gemm(st, PR, 64, FR, 64, UR, 64, 64, 64, 64, 1.f, 0, 0);
  gemm(st, PI_, 64, FI, 64, UR, 64, 64, 64, 64, 1.f, 0, 1);
  gemm(st, PI_, 64, FR, 64, UI, 64, 64, 64, 64, 1.f, 0, 0);
  gemm(st, PR, 64, FI, 64, UI, 64, 64, 64, 64, -1.f, 0, 1);

  k_mag2<<<16, 256, 0, st>>>(UR, UI, SAL);
  k_gauss_blur<<<16, 256, 0, st>>>(SAL, G9, BLR);
  k_rownorm<<<64, 64, 0, st>>>(BLR, NRM);

  // linear upsample 64->256 : Sout = LLIN * NRM * LLIN^T
  gemm(st, LLIN, 64, NRM, 64, UPT, 64, 256, 64, 64, 1.f, 0, 0);
  gemm(st, UPT, 64, LLIN, 64, Sout, 256, 256, 256, 64, 1.f, 1, 0);
}

extern "C" void kernel_launch(void* const* d_in, const int* in_sizes, int n_in,
                              void* d_out, int out_size, void* d_ws, size_t ws_size,
                              hipStream_t stream)
{
  (void)in_sizes; (void)n_in; (void)out_size; (void)ws_size;
  const float* x = (const float*)d_in[0];
  const float* y = (const float*)d_in[1];
  float* out = (float*)d_out;
  float* W = (float*)d_ws;

  float* Y1   = W + 0;
  float* I1   = W + 65536;
  float* Q1   = W + 131072;
  float* Y2   = W + 196608;
  float* I2   = W + 262144;
  float* Q2   = W + 327680;
  float* G1   = W + 393216;
  float* G2   = W + 458752;
  float* S1   = W + 524288;
  float* S2   = W + 589824;
  float* ACUB = W + 655360;    // 64x256
  float* LLIN = W + 671744;    // 256x64
  float* FR   = W + 688128;    // 64x64
  float* FI   = W + 692224;    // 64x64
  float* G9   = W + 696320;    // 16
  float* SCR  = W + 696336;    // 98304 floats saliency scratch
  float* PN   = W + 794640;    // 256 partial numerators
  float* PD   = W + 794896;    // 256 partial denominators
  // total: 795152 floats = ~3.2 MB of d_ws

  // Constant operators
  k_fill_dft<<<16, 256, 0, stream>>>(FR, FI);
  k_fill_cubic<<<64, 256, 0, stream>>>(ACUB);
  k_fill_linear<<<256, 64, 0, stream>>>(LLIN);
  k_fill_gauss9<<<1, 1, 0, stream>>>(G9);

  // Stage 1: the only bandwidth-relevant pass
  k_down_yiq<<<256, 256, 0, stream>>>(x, y, Y1, I1, Q1, Y2, I2, Q2);

  // Saliency for both Y maps (tiny WMMA-driven pipelines)
  run_saliency(stream, Y1, S1, ACUB, LLIN, FR, FI, G9, SCR);
  run_saliency(stream, Y2, S2, ACUB, LLIN, FR, FI, G9, SCR);

  // Gradient magnitudes
  k_grad<<<256, 256, 0, stream>>>(Y1, G1);
  k_grad<<<256, 256, 0, stream>>>(Y2, G2);

  // Combine + deterministic reduction
  k_combine<<<256, 256, 0, stream>>>(S1, S2, G1, G2, I1, I2, Q1, Q2, PN, PD);
  k_final<<<1, 256, 0, stream>>>(PN, PD, out);
}